// EchoLayer_4810363372880
// MI455X (gfx1250) — compile-verified
//
#include <hip/hip_runtime.h>
#include <hip/hip_bf16.h>

// ---------------------------------------------------------------------------
// EchoLayer for MI455X (gfx1250): 5 x [agg = rm @ e ; shift ; renorm]
// fp32 GEMM via v_wmma_f32_16x16x4_f32; rm (134MB) is L2-resident (192MB L2).
// Staging uses GLOBAL_LOAD_ASYNC_TO_LDS_B128 (ASYNCcnt) -- no VGPR round-trip.
// Diagonal masking cancels algebraically: use full rm + full rowsum.
// ---------------------------------------------------------------------------

typedef float v2f __attribute__((ext_vector_type(2)));
typedef float v8f __attribute__((ext_vector_type(8)));

#define ECHO_D      64
#define ECHO_BM     64          // output rows per workgroup
#define ECHO_KC     64          // K chunk staged in LDS
#define ASTR        68          // LDS stride for A (conflict-free b64 reads, 16B-aligned rows)
#define BSTR        80          // LDS stride for B (half-waves hit disjoint bank sets)
#define USTR        68          // LDS stride for update tile
#define ECHO_DELTA  0.1625f
#define ECHO_FSCALE 0.325f
#define ECHO_EPS    1e-12f

// async global->LDS 16B copy; per-lane addresses; tracked by ASYNCcnt
__device__ __forceinline__ void async_g2l_b128(const void* g, void* l) {
    unsigned loff = (unsigned)(uintptr_t)l;            // generic LDS addr[31:0] == LDS offset
    unsigned long long ga = (unsigned long long)(uintptr_t)g;
    asm volatile("global_load_async_to_lds_b128 %0, %1, off"
                 :: "v"(loff), "v"(ga) : "memory");
}
__device__ __forceinline__ void wait_async0() {
    asm volatile("s_wait_asynccnt 0x0" ::: "memory");
}

// --- kernel 1: refined = l2norm(embeddings) ; one wave32 per token ----------
__global__ __launch_bounds__(256) void echo_prep_norm(const float* __restrict__ emb,
                                                      float* __restrict__ e0,
                                                      int n_tok) {
    int token = blockIdx.x * 8 + (threadIdx.x >> 5);
    int lane  = threadIdx.x & 31;
    if (token >= n_tok) return;
    const float* src = emb + (size_t)token * ECHO_D;
    v2f x = *(const v2f*)(src + 2 * lane);
    float s = x.x * x.x + x.y * x.y;
    s += __shfl_xor(s, 1);
    s += __shfl_xor(s, 2);
    s += __shfl_xor(s, 4);
    s += __shfl_xor(s, 8);
    s += __shfl_xor(s, 16);
    float inv = 1.0f / fmaxf(sqrtf(s), ECHO_EPS);
    v2f y = { x.x * inv, x.y * inv };
    *(v2f*)(e0 + (size_t)token * ECHO_D + 2 * lane) = y;
}

// --- kernel 2: rowsum[i] = sum_j rm[i][j] (full row; diagonal cancels) ------
__global__ __launch_bounds__(256) void echo_rowsum(const float* __restrict__ rm,
                                                   float* __restrict__ rs,
                                                   int n_rows, int N) {
    int row  = blockIdx.x * 8 + (threadIdx.x >> 5);
    int lane = threadIdx.x & 31;
    if (row >= n_rows) return;
    const float* src = rm + (size_t)row * N;
    float s = 0.0f;
    for (int j = 0; j < N; j += 128) {
        float4 v = *(const float4*)(src + j + lane * 4);
        s += v.x + v.y + v.z + v.w;
    }
    s += __shfl_xor(s, 1);
    s += __shfl_xor(s, 2);
    s += __shfl_xor(s, 4);
    s += __shfl_xor(s, 8);
    s += __shfl_xor(s, 16);
    if (lane == 0) rs[row] = s;
}

// --- kernel 3: one echo layer: agg = rm@e, shift, renorm --------------------
__global__ __launch_bounds__(256) void echo_layer(const float* __restrict__ rm,
                                                  const float* __restrict__ ein,
                                                  float* __restrict__ eout,
                                                  const float* __restrict__ rowsum,
                                                  const float* __restrict__ freq,
                                                  int layer, int N) {
    __shared__ float lA[ECHO_BM * ASTR];
    __shared__ float lB[ECHO_KC * BSTR];
    __shared__ float lU[ECHO_BM * USTR];

    const int b    = blockIdx.y;
    const int row0 = blockIdx.x * ECHO_BM;
    const int t    = threadIdx.x;
    const int wave = t >> 5;
    const int lane = t & 31;
    const int half = lane >> 4;         // 0 or 1 (lane group)
    const int l16  = lane & 15;

    const float* RM = rm  + (size_t)b * N * N;
    const float* E  = ein + (size_t)b * N * ECHO_D;

    const int rt = wave & 3;            // row-tile 0..3 (16 rows each)
    const int cb = (wave >> 2) * 32;    // col base: 0 or 32 (two 16-col tiles)

    v8f acc0 = {};                      // cols cb+0  .. cb+15
    v8f acc1 = {};                      // cols cb+16 .. cb+31

    const int sr = t >> 2;              // staging: row 0..63
    const int sc = (t & 3) * 16;        // staging: col group (16 floats)

    for (int k0 = 0; k0 < N; k0 += ECHO_KC) {
        // async-stage A chunk (64 rows x 64 K) and B chunk (64 tokens x 64 cols)
        {
            const float* srcA = RM + (size_t)(row0 + sr) * N + k0 + sc;
            const float* srcB = E  + (size_t)(k0 + sr) * ECHO_D + sc;
            float* dstA = &lA[sr * ASTR + sc];
            float* dstB = &lB[sr * BSTR + sc];
#pragma unroll
            for (int i = 0; i < 4; ++i)
                async_g2l_b128(srcA + 4 * i, dstA + 4 * i);
#pragma unroll
            for (int i = 0; i < 4; ++i)
                async_g2l_b128(srcB + 4 * i, dstB + 4 * i);
            if (k0 + ECHO_KC < N)       // global_prefetch_b8 of next A chunk
                __builtin_prefetch(RM + (size_t)(row0 + sr) * N + k0 + ECHO_KC + sc, 0, 3);
        }
        wait_async0();                  // ASYNCcnt==0: LDS writes landed
        __syncthreads();

#pragma unroll
        for (int kk = 0; kk < ECHO_KC; kk += 4) {
            // A 16x4: lane m=l16, VGPR0=K(kk+2*half), VGPR1=K(kk+2*half+1)
            v2f a = *(const v2f*)&lA[(16 * rt + l16) * ASTR + kk + 2 * half];
            // B 4x16: VGPR0 = row kk+2*half (col=l16), VGPR1 = row kk+2*half+1
            v2f b0, b1;
            b0.x = lB[(kk + 2 * half)     * BSTR + cb + l16];
            b0.y = lB[(kk + 2 * half + 1) * BSTR + cb + l16];
            b1.x = lB[(kk + 2 * half)     * BSTR + cb + 16 + l16];
            b1.y = lB[(kk + 2 * half + 1) * BSTR + cb + 16 + l16];
            acc0 = __builtin_amdgcn_wmma_f32_16x16x4_f32(false, a, false, b0,
                                                         (short)0, acc0, false, false);
            acc1 = __builtin_amdgcn_wmma_f32_16x16x4_f32(false, a, false, b1,
                                                         (short)0, acc1, false, false);
        }
        __syncthreads();
    }

    // epilogue: u = e + DELTA*(e*rowsum - agg)*active, staged to LDS
#pragma unroll
    for (int v = 0; v < 8; ++v) {
        int rloc = 16 * rt + v + 8 * half;          // C layout: VGPR v -> M = v + 8*half
        int R    = row0 + rloc;
        float rs = rowsum[(size_t)b * N + R];
        float fr = freq[(size_t)b * N + R];
        float el = fminf(fmaxf(1.0f + floorf(fr * ECHO_FSCALE), 1.0f), 5.0f);
        float act = ((float)layer < el) ? 1.0f : 0.0f;
        float e0 = E[(size_t)R * ECHO_D + cb + l16];
        float e1 = E[(size_t)R * ECHO_D + cb + 16 + l16];
        float u0 = e0 + ECHO_DELTA * (e0 * rs - acc0[v]) * act;
        float u1 = e1 + ECHO_DELTA * (e1 * rs - acc1[v]) * act;
        lU[rloc * USTR + cb + l16]      = u0;
        lU[rloc * USTR + cb + 16 + l16] = u1;
    }
    __syncthreads();

    // cooperative row renorm: 4 threads per row, 16 cols each
    {
        int r = t >> 2;
        int c = (t & 3) * 16;
        float vals[16];
        float s = 0.0f;
#pragma unroll
        for (int i = 0; i < 16; ++i) {
            vals[i] = lU[r * USTR + c + i];
            s += vals[i] * vals[i];
        }
        s += __shfl_xor(s, 1);
        s += __shfl_xor(s, 2);
        float inv = 1.0f / fmaxf(sqrtf(s), ECHO_EPS);
        float* dst = eout + (size_t)b * N * ECHO_D + (size_t)(row0 + r) * ECHO_D + c;
#pragma unroll
        for (int i = 0; i < 16; ++i) dst[i] = vals[i] * inv;
    }
}

// ---------------------------------------------------------------------------
extern "C" void kernel_launch(void* const* d_in, const int* in_sizes, int n_in,
                              void* d_out, int out_size, void* d_ws, size_t ws_size,
                              hipStream_t stream) {
    const float* emb  = (const float*)d_in[0];   // [B,N,D] fp32
    const float* freq = (const float*)d_in[1];   // [B,N]   fp32
    const float* rm   = (const float*)d_in[2];   // [B,N,N] fp32

    const int BN = in_sizes[1];                  // B*N = 16384
    const int N  = in_sizes[2] / in_sizes[1];    // 2048
    const int B  = BN / N;                       // 8

    float* eA = (float*)d_ws;                                        // ping buffer
    float* rs = (float*)((char*)d_ws + (size_t)BN * ECHO_D * 4);     // rowsums
    float* eB = (float*)d_out;                                       // pong buffer

    dim3 blk(256);
    echo_prep_norm<<<dim3((BN + 7) / 8), blk, 0, stream>>>(emb, eA, BN);
    echo_rowsum  <<<dim3((BN + 7) / 8), blk, 0, stream>>>(rm, rs, BN, N);

    dim3 grid(N / ECHO_BM, B);
    // 5 layers ping-ponging ws <-> d_out; final result lands in d_out.
    echo_layer<<<grid, blk, 0, stream>>>(rm, eA, eB, rs, freq, 0, N);
    echo_layer<<<grid, blk, 0, stream>>>(rm, eB, eA, rs, freq, 1, N);
    echo_layer<<<grid, blk, 0, stream>>>(rm, eA, eB, rs, freq, 2, N);
    echo_layer<<<grid, blk, 0, stream>>>(rm, eB, eA, rs, freq, 3, N);
    echo_layer<<<grid, blk, 0, stream>>>(rm, eA, eB, rs, freq, 4, N);
}